// HouseHolder_30176440221861
// MI455X (gfx1250) — compile-verified
//
#include <hip/hip_runtime.h>
#include <math.h>

typedef __attribute__((ext_vector_type(2))) float v2f;
typedef __attribute__((ext_vector_type(4))) float v4f;
typedef __attribute__((ext_vector_type(8))) float v8f;

#define DIM 1024
#define NVEC 8

// ---------------------------------------------------------------------------
// Kernel 1: normalize vectors, build compact-WY factor T, emit
//   ws[0      .. 8*1024)  : Yt[j][k]  where Y = V * T   (V columns = v_j)
//   ws[8*1024 .. 16*1024) : Vn[j][k]  normalized vectors
// P = (I-2v0v0^T)...(I-2v7v7^T) = I - V T V^T, so out = X - (X*Y) * Vn^T.
// ---------------------------------------------------------------------------
__global__ __launch_bounds__(256) void hh_prep(const float* __restrict__ vecs,
                                               float* __restrict__ ws) {
  __shared__ float sV[NVEC * DIM];
  __shared__ float sRed[256];
  __shared__ float sG[NVEC][NVEC];
  __shared__ float sT[NVEC][NVEC];
  const int tid = threadIdx.x;

  // normalize: v / max(||v||, 1e-12)  (matches reference)
  for (int j = 0; j < NVEC; ++j) {
    float p = 0.f;
    for (int i = tid; i < DIM; i += 256) {
      float v = vecs[j * DIM + i];
      p += v * v;
    }
    sRed[tid] = p;
    __syncthreads();
    for (int s = 128; s > 0; s >>= 1) {
      if (tid < s) sRed[tid] += sRed[tid + s];
      __syncthreads();
    }
    float rinv = 1.0f / fmaxf(sqrtf(sRed[0]), 1e-12f);
    __syncthreads();
    for (int i = tid; i < DIM; i += 256)
      sV[j * DIM + i] = vecs[j * DIM + i] * rinv;
    __syncthreads();
  }

  // Gram upper triangle G[m][k] = v_m . v_k (m < k)
  for (int k = 1; k < NVEC; ++k) {
    for (int m = 0; m < k; ++m) {
      float p = 0.f;
      for (int i = tid; i < DIM; i += 256)
        p += sV[m * DIM + i] * sV[k * DIM + i];
      sRed[tid] = p;
      __syncthreads();
      for (int s = 128; s > 0; s >>= 1) {
        if (tid < s) sRed[tid] += sRed[tid + s];
        __syncthreads();
      }
      if (tid == 0) sG[m][k] = sRed[0];
      __syncthreads();
    }
  }

  // T recurrence: T_k = [[T_{k-1}, -2 T_{k-1} g_k],[0, 2]]
  if (tid == 0) {
    for (int i = 0; i < NVEC; ++i)
      for (int j = 0; j < NVEC; ++j) sT[i][j] = 0.f;
    sT[0][0] = 2.f;
    for (int k = 1; k < NVEC; ++k) {
      for (int i = 0; i < k; ++i) {
        float z = 0.f;
        for (int m = i; m < k; ++m) z += sT[i][m] * sG[m][k];
        sT[i][k] = -2.f * z;
      }
      sT[k][k] = 2.f;
    }
  }
  __syncthreads();

  // Yt[j][k] = sum_i T[i][j] * Vn[i][k] ; also store Vn
  for (int kk = tid; kk < DIM; kk += 256) {
    for (int j = 0; j < NVEC; ++j) {
      float y = 0.f;
      for (int i = 0; i < NVEC; ++i) y += sT[i][j] * sV[i * DIM + kk];
      ws[j * DIM + kk] = y;
      ws[NVEC * DIM + j * DIM + kk] = sV[j * DIM + kk];
    }
  }
}

// ---------------------------------------------------------------------------
// Kernel 2: out = X - (X*Y) * Vn^T, one 16-row tile per wave, all math on
// v_wmma_f32_16x16x4_f32 (wave32 layouts per CDNA5 ISA 7.12.2):
//   A 16x4 : lane<16 holds row=lane k={0,1}; lane>=16 holds row=lane-16 k={2,3}
//   B 4x16 : vgpr0 = rows k=2*hi, vgpr1 = rows k=2*hi+1, col = lane&15
//   C 16x16: vgpr r = row r + 8*hi, col = lane&15
// ---------------------------------------------------------------------------
__global__ __launch_bounds__(256) void hh_apply(const float* __restrict__ X,
                                                const float* __restrict__ ws,
                                                float* __restrict__ out) {
  __shared__ float sYtp[16 * DIM];      // Yt padded with 8 zero rows -> N-pad
  __shared__ float sVn[NVEC * DIM];
  __shared__ float sP[8][16 * 16];      // per-wave -P staging (C->A transpose)

  const int tid = threadIdx.x;

  // cooperative LDS fill (b128 traffic)
  {
    const v4f* src = (const v4f*)ws;
    v4f* dY = (v4f*)sYtp;
    v4f* dV = (v4f*)sVn;
    const int n4 = NVEC * DIM / 4;      // 2048
    for (int i = tid; i < n4; i += 256) {
      dY[i] = src[i];
      dV[i] = src[n4 + i];
    }
    v4f z = {0.f, 0.f, 0.f, 0.f};
    for (int i = tid; i < n4; i += 256) dY[n4 + i] = z;  // zero rows 8..15
  }
  __syncthreads();

  const int lane = tid & 31;
  const int wid  = tid >> 5;
  const int n    = lane & 15;
  const int hi   = lane >> 4;
  const int rowbase = (blockIdx.x * 8 + wid) * 16;

  // ---- Phase 1: P(16x16, cols 8..15 = 0) = Xtile(16x1024) * Y(1024x16) ----
  v8f acc = {};
  const float* aptr = X + (size_t)(rowbase + n) * DIM + 2 * hi;
  const float* bptr = sYtp + n * DIM + 2 * hi;
#pragma unroll 8
  for (int k0 = 0; k0 < DIM; k0 += 4) {
    v2f a = *(const v2f*)(aptr + k0);   // global_load_b64, imm offset k0*4
    v2f b = *(const v2f*)(bptr + k0);   // ds_load_b64,     imm offset k0*4
    acc = __builtin_amdgcn_wmma_f32_16x16x4_f32(false, a, false, b,
                                                (short)0, acc, false, false);
  }

  // ---- stage -P through LDS: C-layout -> A-layout transpose ----
  float* pst = &sP[wid][0];
#pragma unroll
  for (int r = 0; r < 8; ++r)
    pst[(r + 8 * hi) * 16 + n] = -acc[r];
  __syncthreads();

  v2f a1 = *(const v2f*)(pst + n * 16 + 2 * hi);       // A rows=n, k=2hi,2hi+1
  v2f a2 = *(const v2f*)(pst + n * 16 + 4 + 2 * hi);   // k = 4+2hi, 5+2hi

  // ---- Phase 2: out tile = Xtile + (-P)(16x8) * Vn^T(8x1024) ----
  const size_t cbase = (size_t)(rowbase + 8 * hi) * DIM + n;
  for (int cb = 0; cb < 64; ++cb) {
    const int coff = cb * 16;
    v8f c;
#pragma unroll
    for (int r = 0; r < 8; ++r)
      c[r] = X[cbase + (size_t)r * DIM + coff];        // C-layout accumulator
    v2f b1, b2;
    b1[0] = sVn[(2 * hi + 0) * DIM + coff + n];        // B rows j=0..3
    b1[1] = sVn[(2 * hi + 1) * DIM + coff + n];
    b2[0] = sVn[(2 * hi + 4) * DIM + coff + n];        // B rows j=4..7
    b2[1] = sVn[(2 * hi + 5) * DIM + coff + n];
    c = __builtin_amdgcn_wmma_f32_16x16x4_f32(false, a1, false, b1,
                                              (short)0, c, false, false);
    c = __builtin_amdgcn_wmma_f32_16x16x4_f32(false, a2, false, b2,
                                              (short)0, c, false, false);
#pragma unroll
    for (int r = 0; r < 8; ++r)                        // NT store: don't pollute L2
      __builtin_nontemporal_store(c[r], &out[cbase + (size_t)r * DIM + coff]);
  }
}

extern "C" void kernel_launch(void* const* d_in, const int* in_sizes, int n_in,
                              void* d_out, int out_size, void* d_ws, size_t ws_size,
                              hipStream_t stream) {
  const float* X    = (const float*)d_in[0];
  const float* vecs = (const float*)d_in[1];
  float* outp = (float*)d_out;
  float* ws   = (float*)d_ws;

  const int rows = in_sizes[0] / DIM;        // 65536
  const int blocks = rows / (8 * 16);        // 8 waves * 16 rows per block

  hh_prep<<<1, 256, 0, stream>>>(vecs, ws);
  hh_apply<<<blocks, 256, 0, stream>>>(X, ws, outp);
}